// GeneralConv_86277303042050
// MI455X (gfx1250) — compile-verified
//
#include <hip/hip_runtime.h>
#include <hip/hip_bf16.h>

#define N_NODES 50000
#define N_EDGES 640000
#define F_IN    128
#define HEADS   4
#define C_OUT   64
#define HC      (HEADS * C_OUT)   // 256

typedef __attribute__((ext_vector_type(2))) float v2f;
typedef __attribute__((ext_vector_type(8))) float v8f;

// ---------------------------------------------------------------------------
// WMMA f32 GEMM: D[M,Ncols] = A[M,K] @ W[K,Ncols] + bias.
// One wave computes one 16x16 output tile with V_WMMA_F32_16X16X4_F32
// (K-loop of 32 WMMAs for K=128).  M=50000 = 3125*16, K=128, Ncols in {64,256}
// all divide evenly -> no guards, EXEC stays all-ones for WMMA.
//
// A fragment (16x4 f32): lanes 0-15 hold M=lane, K={0,1}; lanes 16-31 hold
// M=lane-16, K={2,3}; one v2f (contiguous -> single b64 load).
// B fragment (4x16 f32): lanes 0-15 hold N=lane, K={0,1}; lanes 16-31 K={2,3}.
// C/D: VGPR i = row (lane>=16 ? 8+i : i), col = lane&15.
// ---------------------------------------------------------------------------
__global__ void gat_wmma_gemm_f32(const float* __restrict__ A,
                                  const float* __restrict__ W,
                                  const float* __restrict__ bias,
                                  float* __restrict__ D,
                                  int K, int Ncols) {
    const int lane = threadIdx.x & 31;
    const int wave = threadIdx.x >> 5;
    const int wavesPerBlock = blockDim.x >> 5;

    const int m0 = blockIdx.x * 16;                                 // row tile
    const int n0 = (blockIdx.y * wavesPerBlock + wave) * 16;        // col tile

    const int mA = lane & 15;
    const int kk = (lane >> 4) * 2;      // 0 or 2
    const int nB = lane & 15;

    const float* __restrict__ arow = A + (size_t)(m0 + mA) * K;     // 512B aligned
    const float* __restrict__ wcol = W + n0 + nB;

    v8f acc = {};
    for (int k = 0; k < K; k += 4) {
        v2f a = *(const v2f*)(arow + k + kk);                       // b64 load
        v2f b;
        b.x = wcol[(size_t)(k + kk)     * Ncols];
        b.y = wcol[(size_t)(k + kk + 1) * Ncols];
        acc = __builtin_amdgcn_wmma_f32_16x16x4_f32(
                  false, a, false, b, (short)0, acc, false, false);
    }

    const float bv = bias[n0 + nB];
    const int mBase = (lane >> 4) * 8;
#pragma unroll
    for (int i = 0; i < 8; ++i) {
        D[(size_t)(m0 + mBase + i) * Ncols + n0 + nB] = acc[i] + bv;
    }
}

// ---------------------------------------------------------------------------
// Per-node attention pre-logit:  t[n,h] = leaky_relu(sum_c proj[n,h,c]*alpha[h,c])
// One wave per node; lane loads 8 contiguous channels (8 | 64 so each lane's
// chunk lies in one head); butterfly-reduce within groups of 8 lanes.
// ---------------------------------------------------------------------------
__global__ void gat_node_logits(const float* __restrict__ proj,
                                const float* __restrict__ alpha,
                                float* __restrict__ t, int N) {
    const int n = blockIdx.x * (blockDim.x >> 5) + (threadIdx.x >> 5);
    if (n >= N) return;
    const int lane = threadIdx.x & 31;
    const float* __restrict__ p  = proj  + (size_t)n * HC + lane * 8;
    const float* __restrict__ al = alpha + lane * 8;
    float s = 0.f;
#pragma unroll
    for (int i = 0; i < 8; ++i) s += p[i] * al[i];
    s += __shfl_xor(s, 1, 32);
    s += __shfl_xor(s, 2, 32);
    s += __shfl_xor(s, 4, 32);
    if ((lane & 7) == 0) {
        const int h = lane >> 3;
        t[n * HEADS + h] = (s > 0.f) ? s : 0.01f * s;   // leaky_relu slope 0.01
    }
}

// Monotonic order-preserving uint encoding of float (for atomicMax on f32).
__device__ __forceinline__ unsigned ford(float f) {
    unsigned u = __float_as_uint(f);
    return (u & 0x80000000u) ? ~u : (u | 0x80000000u);
}
__device__ __forceinline__ float funord(unsigned u) {
    return __uint_as_float((u & 0x80000000u) ? (u & 0x7FFFFFFFu) : ~u);
}

__global__ void gat_init(unsigned* __restrict__ segmax,
                         float* __restrict__ denom, int n) {
    const int i = blockIdx.x * blockDim.x + threadIdx.x;
    if (i < n) { segmax[i] = 0u; denom[i] = 0.0f; }
}

// Pass 1 over edges: segment max of logits per (receiver, head).
__global__ void gat_edge_max(const float* __restrict__ t,
                             const int* __restrict__ senders,
                             const int* __restrict__ receivers,
                             unsigned* __restrict__ segmax, int E4) {
    const int i = blockIdx.x * blockDim.x + threadIdx.x;
    if (i >= E4) return;
    const int e = i >> 2, h = i & 3;
    const int s = senders[e], r = receivers[e];
    atomicMax(segmax + r * HEADS + h, ford(t[s * HEADS + h]));
}

// Pass 2: e = exp(logit - segmax[recv]); cache per-edge, atomic-add denom.
__global__ void gat_edge_exp(const float* __restrict__ t,
                             const unsigned* __restrict__ segmax,
                             const int* __restrict__ senders,
                             const int* __restrict__ receivers,
                             float* __restrict__ evals,
                             float* __restrict__ denom, int E4) {
    const int i = blockIdx.x * blockDim.x + threadIdx.x;
    if (i >= E4) return;
    const int e = i >> 2, h = i & 3;
    const int s = senders[e], r = receivers[e];
    const float m  = funord(segmax[r * HEADS + h]);
    const float ev = __expf(t[s * HEADS + h] - m);
    evals[i] = ev;
    atomicAdd(denom + r * HEADS + h, ev);
}

__global__ void gat_recip(const float* __restrict__ denom,
                          float* __restrict__ rdenom, int n) {
    const int i = blockIdx.x * blockDim.x + threadIdx.x;
    if (i < n) rdenom[i] = 1.0f / denom[i];
}

// Pass 3: out[r,c] += (1/H) * sum_h att[e,h] * proj[s,h,c]
// 64 threads per edge (one per output channel); head-mean folded in so only
// 64 float atomics per edge.  proj + out are L2-resident.
__global__ void gat_edge_aggregate(const float* __restrict__ proj,
                                   const float* __restrict__ evals,
                                   const float* __restrict__ rdenom,
                                   const int* __restrict__ senders,
                                   const int* __restrict__ receivers,
                                   float* __restrict__ out, int E) {
    const int gid = blockIdx.x * blockDim.x + threadIdx.x;
    const int e = gid >> 6;
    if (e >= E) return;
    const int c = gid & 63;
    const int s = senders[e], r = receivers[e];
    const float* __restrict__ pe = evals  + (size_t)e * HEADS;
    const float* __restrict__ pd = rdenom + (size_t)r * HEADS;
    const float* __restrict__ pp = proj   + (size_t)s * HC + c;
    float acc = 0.f;
#pragma unroll
    for (int h = 0; h < HEADS; ++h)
        acc += (pe[h] * pd[h]) * pp[h * C_OUT];
    atomicAdd(out + (size_t)r * C_OUT + c, 0.25f * acc);
}

extern "C" void kernel_launch(void* const* d_in, const int* in_sizes, int n_in,
                              void* d_out, int out_size, void* d_ws, size_t ws_size,
                              hipStream_t stream) {
    const float* nodes   = (const float*)d_in[0];   // [N,128]
    const float* W_msg   = (const float*)d_in[1];   // [128,256]
    const float* b_msg   = (const float*)d_in[2];   // [256]
    const float* alpha   = (const float*)d_in[3];   // [256] ([1,H,C] flat)
    const float* W_skip  = (const float*)d_in[4];   // [128,64]
    const float* b_skip  = (const float*)d_in[5];   // [64]
    const int*   senders   = (const int*)d_in[6];   // [E]
    const int*   receivers = (const int*)d_in[7];   // [E]
    float* out = (float*)d_out;                     // [N,64]

    // Workspace carving (all offsets multiples of 256B)
    char* ws = (char*)d_ws;
    float*    proj    = (float*)(ws);                          // 50000*256*4 = 51,200,000
    float*    t       = (float*)(ws + 51200000);               // 50000*4*4   =    800,000
    unsigned* segmax  = (unsigned*)(ws + 52000000);            //    800,000
    float*    denom   = (float*)(ws + 52800000);               //    800,000
    float*    rdenom  = (float*)(ws + 53600000);               //    800,000
    float*    evals   = (float*)(ws + 54400000);               // 640000*4*4  = 10,240,000

    const int rowTiles = N_NODES / 16;   // 3125

    // 1) skip GEMM -> d_out (out starts as skip; scatter adds on top)
    //    4 waves/block -> 4 col tiles = 64 cols
    gat_wmma_gemm_f32<<<dim3(rowTiles, 1), 128, 0, stream>>>(
        nodes, W_skip, b_skip, out, F_IN, C_OUT);

    // 2) proj GEMM -> ws   8 waves/block * grid.y=2 -> 16 col tiles = 256 cols
    gat_wmma_gemm_f32<<<dim3(rowTiles, 2), 256, 0, stream>>>(
        nodes, W_msg, b_msg, proj, F_IN, HC);

    // 3) per-node attention logits (+ leaky_relu)
    gat_node_logits<<<(N_NODES + 7) / 8, 256, 0, stream>>>(proj, alpha, t, N_NODES);

    // 4) init segment max / denom
    const int NH = N_NODES * HEADS;
    gat_init<<<(NH + 255) / 256, 256, 0, stream>>>(segmax, denom, NH);

    // 5) segment max over edges
    const int E4 = N_EDGES * HEADS;
    gat_edge_max<<<(E4 + 255) / 256, 256, 0, stream>>>(t, senders, receivers, segmax, E4);

    // 6) exp + denominator
    gat_edge_exp<<<(E4 + 255) / 256, 256, 0, stream>>>(t, segmax, senders, receivers,
                                                       evals, denom, E4);

    // 7) reciprocal of denominators
    gat_recip<<<(NH + 255) / 256, 256, 0, stream>>>(denom, rdenom, NH);

    // 8) weighted scatter-add (head-mean folded)
    const long long aggThreads = (long long)N_EDGES * 64;
    gat_edge_aggregate<<<(int)((aggThreads + 255) / 256), 256, 0, stream>>>(
        proj, evals, rdenom, senders, receivers, out, N_EDGES);
}